// MentionRepresentation_1795296329896
// MI455X (gfx1250) — compile-verified
//
#include <hip/hip_runtime.h>
#include <stdint.h>

// Problem constants (fixed by the reference's setup_inputs).
#define Bn 2
#define Mn 128
#define Sn 1024
#define Hn 768

#define MT 16                 // mentions per block
#define HT 128                // h columns per block
#define SK 32                 // s positions per chunk (= one 32-bit mask word)
#define SB 8                  // S splits across blocks
#define S_PER_BLOCK (Sn / SB) // 128
#define CHUNKS (S_PER_BLOCK / SK) // 4
#define NEGV (-1e30f)

__global__ __launch_bounds__(256)
void mention_out_init(float* __restrict__ out, int n) {
    int i = blockIdx.x * 256 + threadIdx.x;
    if (i < n) out[i] = NEGV;
}

__global__ __launch_bounds__(256)
void mention_maxpool(const float* __restrict__ h,
                     const int* __restrict__ mask,
                     float* __restrict__ out) {
    // Double-buffered h tile: 2 x (32 s x 128 cols) fp32 = 32 KB.
    __shared__ __align__(16) float hbuf[2][SK * HT];
    // Packed mention masks: MT rows x CHUNKS 32-bit words.
    __shared__ uint32_t pm[MT * CHUNKS];

    const int tid  = threadIdx.x;
    const int lane = tid & 31;
    const int wv   = tid >> 5;          // wave id 0..7
    const int hb   = blockIdx.x;        // 0..5   H tile
    const int mb   = blockIdx.y;        // 0..7   M tile
    const int bz   = blockIdx.z;        // 0..15  (b, s-split)
    const int b    = bz >> 3;
    const int sblk = bz & (SB - 1);
    const int h0 = hb * HT;
    const int m0 = mb * MT;
    const int s0 = sblk * S_PER_BLOCK;

    // Issue one chunk of async global->LDS copies.
    // Chunk = 32 rows x 128 floats; wave wv loads rows wv*4 .. wv*4+3,
    // each lane moves 16 bytes -> 4 async b128 ops per wave per chunk.
    auto issue = [&](int ck, int bsel) {
        #pragma unroll
        for (int i = 0; i < 4; ++i) {
            int r = wv * 4 + i;
            int s = s0 + ck * SK + r;
            const float* gp = h + ((size_t)(b * Sn + s)) * Hn + h0 + lane * 4;
            uint32_t lds = (uint32_t)(uintptr_t)(&hbuf[bsel][r * HT + lane * 4]);
            asm volatile("global_load_async_to_lds_b128 %0, %1, off"
                         :: "v"(lds), "v"((uint64_t)(uintptr_t)gp)
                         : "memory");
        }
    };

    // Kick chunk 0 immediately so the async engine overlaps mask packing.
    issue(0, 0);

    // Pack mention masks into bitfields with wave32 ballot:
    // word idx -> (m_local = idx/CHUNKS, chunk = idx%CHUNKS), bit sl = s offset.
    for (int idx = wv; idx < MT * CHUNKS; idx += 8) {
        int ml = idx / CHUNKS;
        int ck = idx % CHUNKS;
        int s  = s0 + ck * SK + lane;
        int v  = mask[((size_t)(b * Mn + m0 + ml)) * Sn + s];
        uint32_t bits = __builtin_amdgcn_ballot_w32(v != 0);
        if (lane == 0) pm[idx] = bits;
    }
    __syncthreads();

    // Each thread owns one column c for 8 mentions (mg selects upper/lower 8).
    const int c  = tid & (HT - 1);
    const int mg = tid >> 7; // 0 or 1, constant within a wave
    float acc[8];
    #pragma unroll
    for (int j = 0; j < 8; ++j) acc[j] = NEGV;

    #pragma unroll 1
    for (int ck = 0; ck < CHUNKS; ++ck) {
        const int cur = ck & 1;
        if (ck + 1 < CHUNKS) {
            issue(ck + 1, cur ^ 1);
            // 4 loads of the *next* chunk may stay in flight; ours are done.
            asm volatile("s_wait_asynccnt 0x4" ::: "memory");
        } else {
            asm volatile("s_wait_asynccnt 0x0" ::: "memory");
        }
        __syncthreads(); // chunk ck visible to all waves

        // Mask words are wave-uniform (mg uniform per wave) -> SGPRs.
        uint32_t sbits[8];
        #pragma unroll
        for (int j = 0; j < 8; ++j)
            sbits[j] = (uint32_t)__builtin_amdgcn_readfirstlane(
                (int)pm[(mg * 8 + j) * CHUNKS + ck]);

        #pragma unroll
        for (int sl = 0; sl < SK; ++sl) {
            float hv = hbuf[cur][sl * HT + c]; // conflict-free ds_load_b32
            #pragma unroll
            for (int j = 0; j < 8; ++j) {
                // Scalar-side select of {0, -1e30}; VALU does add+max only.
                float add = ((sbits[j] >> sl) & 1u) ? 0.0f : NEGV;
                acc[j] = fmaxf(acc[j], hv + add);
            }
        }
        __syncthreads(); // all waves done reading before buffer is reloaded
    }

    // Combine S-split partials; out pre-initialized to -1e30.
    #pragma unroll
    for (int j = 0; j < 8; ++j) {
        int m = m0 + mg * 8 + j;
        atomicMax(&out[((size_t)(b * Mn + m)) * Hn + h0 + c], acc[j]);
    }
}

extern "C" void kernel_launch(void* const* d_in, const int* in_sizes, int n_in,
                              void* d_out, int out_size, void* d_ws, size_t ws_size,
                              hipStream_t stream) {
    const float* h    = (const float*)d_in[0];
    const int*   mask = (const int*)d_in[1];
    float*       out  = (float*)d_out;

    const int n_out = Bn * Mn * Hn; // 196608
    mention_out_init<<<(n_out + 255) / 256, 256, 0, stream>>>(out, n_out);

    dim3 grid(Hn / HT, Mn / MT, Bn * SB); // 6 x 8 x 16 = 768 blocks
    mention_maxpool<<<grid, 256, 0, stream>>>(h, mask, out);
}